// TorchNERFBuilder_Oxygen_59450937312103
// MI455X (gfx1250) — compile-verified
//
#include <hip/hip_runtime.h>
#include <stdint.h>
#include <stddef.h>

// MI455X / gfx1250 NeRF backbone builder.
// One wave32 per chain (B=8192 chains, L=512 residues), 16 residues per lane.
//   Pass 0: async-stage the chain's 4x512 torsions into LDS
//           (global_load_async_to_lds_b128, ASYNCcnt-tracked, coalesced,
//            4-float-skewed LDS layout -> near-conflict-free ds reads).
//   Pass 1: each lane runs its 16 NeRF steps from the canonical seed triple,
//           extracts its block rigid transform (R|t) — NeRF steps are
//           equivariant under proper rigid motions, so the chain is a prefix
//           product of rigid transforms.
//   Scan:   wave-level Hillis-Steele inclusive scan of (R|t) via __shfl_up
//           (5 rounds x 12 lanespread values), shifted to exclusive prefix.
//   Pass 2: re-run local steps, apply prefix transform, write atoms with
//           non-temporal b128 stores (201 MB write-once stream).
// No WMMA: the matrix work is 3x3 rigid composes (5 per wave); a 16x16 WMMA
// tile would be >90% padding and force f16 on coordinates that accumulate
// over 512 steps. f32 VALU + v_sin/v_cos is the right engine here.

namespace {

constexpr int LRES = 512;      // residues per chain
constexpr int WPB  = 4;        // waves (=chains) per block
constexpr int TPB  = WPB * 32;
constexpr int SROW = 640;      // skewed LDS row stride (max p(511)=635)

typedef float v4f __attribute__((ext_vector_type(4)));

// dbc = -len*cos(angle), k = len*sin(angle)  (precomputed f32)
constexpr float DBC_N  = 0.56630847f, K_N  = 1.21445244f; // C-N  115.0deg 1.34
constexpr float DBC_CA = 0.75195558f, K_CA = 1.25146426f; // N-CA 121.0deg 1.46
constexpr float DBC_C  = 0.50137495f, K_C  = 1.45609861f; // CA-C 109.0deg 1.54
constexpr float DBC_O  = 0.62427229f, K_O  = 1.05980384f; // C-O  120.5deg 1.23

struct F3 { float x, y, z; };

__device__ inline F3 sub3(F3 a, F3 b) { return {a.x - b.x, a.y - b.y, a.z - b.z}; }
__device__ inline F3 crs3(F3 a, F3 b) {
  return {a.y * b.z - a.z * b.y, a.z * b.x - a.x * b.z, a.x * b.y - a.y * b.x};
}
__device__ inline F3 unit3(F3 a) {
  float r = rsqrtf(a.x * a.x + a.y * a.y + a.z * a.z);
  return {a.x * r, a.y * r, a.z * r};
}

// 4-float skew: element j lives at LDS row offset j + 4*(j>>4).
__device__ inline int skew(int j) { return j + ((j >> 4) << 2); }

// Rigid transform, column form: p' = cx*p.x + cy*p.y + cz*p.z + t
struct Xf { F3 cx, cy, cz, t; };

__device__ inline F3 xrot(const Xf& A, F3 p) {
  return {A.cx.x * p.x + A.cy.x * p.y + A.cz.x * p.z,
          A.cx.y * p.x + A.cy.y * p.y + A.cz.y * p.z,
          A.cx.z * p.x + A.cy.z * p.y + A.cz.z * p.z};
}
__device__ inline F3 xapply(const Xf& A, F3 p) {
  F3 r = xrot(A, p);
  return {r.x + A.t.x, r.y + A.t.y, r.z + A.t.z};
}
__device__ inline Xf xcompose(const Xf& A, const Xf& B) {  // (A o B)(p) = A(B(p))
  Xf r;
  r.cx = xrot(A, B.cx);
  r.cy = xrot(A, B.cy);
  r.cz = xrot(A, B.cz);
  r.t  = xapply(A, B.t);
  return r;
}
__device__ inline Xf xshfl_up(const Xf& v, int d) {
  Xf r;
  r.cx.x = __shfl_up(v.cx.x, d, 32); r.cx.y = __shfl_up(v.cx.y, d, 32); r.cx.z = __shfl_up(v.cx.z, d, 32);
  r.cy.x = __shfl_up(v.cy.x, d, 32); r.cy.y = __shfl_up(v.cy.y, d, 32); r.cy.z = __shfl_up(v.cy.z, d, 32);
  r.cz.x = __shfl_up(v.cz.x, d, 32); r.cz.y = __shfl_up(v.cz.y, d, 32); r.cz.z = __shfl_up(v.cz.z, d, 32);
  r.t.x  = __shfl_up(v.t.x,  d, 32); r.t.y  = __shfl_up(v.t.y,  d, 32); r.t.z  = __shfl_up(v.t.z,  d, 32);
  return r;
}

// place_dihedral: matches reference (ab NOT normalized; frame orthonormalized)
__device__ inline F3 place(F3 a, F3 b, F3 c, float dbc, float k, float tau) {
  F3 ab  = sub3(b, a);
  F3 bc  = unit3(sub3(c, b));
  F3 n   = unit3(crs3(ab, bc));
  F3 nbc = crs3(n, bc);
  float st, ct;
  __sincosf(tau, &st, &ct);
  float w1 = k * ct, w2 = k * st;
  return {c.x + dbc * bc.x + w1 * nbc.x + w2 * n.x,
          c.y + dbc * bc.y + w1 * nbc.y + w2 * n.y,
          c.z + dbc * bc.z + w1 * nbc.z + w2 * n.z};
}

__global__ __launch_bounds__(TPB) void nerf_build_kernel(
    const float* __restrict__ phi, const float* __restrict__ psi,
    const float* __restrict__ omg, const float* __restrict__ odh,
    float* __restrict__ out, int B) {
  __shared__ float smem[WPB * 4 * SROW];  // 40 KB: per-wave skewed [4][640]

  const int w    = threadIdx.x >> 5;
  const int lane = threadIdx.x & 31;
  const int b    = blockIdx.x * WPB + w;
  if (b >= B) return;  // wave-private LDS, no barriers -> early exit safe

  const int wbase = w * (4 * SROW);

  // ---- Pass 0: async-stage torsions into LDS (coalesced b128 per lane) ----
  // smem sits at LDS offset 0 (sole shared object); the async op's LDS
  // operand is a per-lane byte offset within the workgroup's LDS allocation.
  // Skew keeps each aligned 16-element group contiguous, so b128 is legal.
  {
    const float* srcs[4] = {phi + (size_t)b * LRES, psi + (size_t)b * LRES,
                            omg + (size_t)b * LRES, odh + (size_t)b * LRES};
#pragma unroll
    for (int a = 0; a < 4; ++a) {
      const float* src = srcs[a];
#pragma unroll
      for (int k = 0; k < 4; ++k) {
        int j = 4 * lane + 128 * k;          // coalesced: wave covers 512B
        const float* gp = src + j;
        unsigned lofs = (unsigned)(wbase + a * SROW + skew(j)) * 4u;
        asm volatile("global_load_async_to_lds_b128 %0, %1, off"
                     :: "v"(lofs), "v"(gp) : "memory");
      }
    }
    asm volatile("s_wait_asynccnt 0x0" ::: "memory");
  }

  // Canonical seed atoms (residue 0)
  const F3 N0  = {17.047f, 14.099f, 3.625f};
  const F3 CA0 = {16.967f, 12.784f, 4.338f};
  const F3 C0  = {15.685f, 12.755f, 5.133f};
  const F3 O0  = {15.268f, 13.825f, 5.594f};

  // Seed frame F0 = (R0, C0), R0 columns u0,v0,w0
  const F3 u0 = unit3(sub3(C0, CA0));
  const F3 w0 = unit3(crs3(sub3(CA0, N0), u0));
  const F3 v0 = crs3(w0, u0);

  const float* sphi = smem + wbase + 0 * SROW;
  const float* spsi = smem + wbase + 1 * SROW;
  const float* somg = smem + wbase + 2 * SROW;
  const float* sodh = smem + wbase + 3 * SROW;

  // Lane l handles residues [16l, 16l+16); step torsion index i = 16l-1+s.
  const int base = lane * 16 - 1;

  // ---- Pass 1: local chain from canonical seed; derive block transform ----
  F3 N = N0, CA = CA0, C = C0;
  for (int s = 0; s < 16; ++s) {
    int i = base + s;
    if (i >= 0) {  // lane 0 skips s==0 (residue 0 is the literal seed)
      float tpsi = spsi[skew(i)], tomg = somg[skew(i)], tphi = sphi[skew(i + 1)];
      F3 Nn  = place(N, CA, C, DBC_N, K_N, tpsi);
      F3 CAn = place(CA, C, Nn, DBC_CA, K_CA, tomg);
      F3 Cn  = place(C, Nn, CAn, DBC_C, K_C, tphi);
      N = Nn; CA = CAn; C = Cn;  // O not needed for the frame
    }
  }

  // T = Frame(end) o Frame(seed)^-1 : R = Re*R0^T, t = Ce - R*C0
  Xf T;
  {
    F3 ue = unit3(sub3(C, CA));
    F3 we = unit3(crs3(sub3(CA, N), ue));
    F3 ve = crs3(we, ue);
    T.cx = {ue.x * u0.x + ve.x * v0.x + we.x * w0.x,
            ue.y * u0.x + ve.y * v0.x + we.y * w0.x,
            ue.z * u0.x + ve.z * v0.x + we.z * w0.x};
    T.cy = {ue.x * u0.y + ve.x * v0.y + we.x * w0.y,
            ue.y * u0.y + ve.y * v0.y + we.y * w0.y,
            ue.z * u0.y + ve.z * v0.y + we.z * w0.y};
    T.cz = {ue.x * u0.z + ve.x * v0.z + we.x * w0.z,
            ue.y * u0.z + ve.y * v0.z + we.y * w0.z,
            ue.z * u0.z + ve.z * v0.z + we.z * w0.z};
    F3 rc0 = xrot(T, C0);
    T.t = {C.x - rc0.x, C.y - rc0.y, C.z - rc0.z};
  }

  // ---- Wave-level scan of rigid transforms (non-commutative, L->R) ----
  Xf S = T;
#pragma unroll
  for (int off = 1; off < 32; off <<= 1) {
    Xf o = xshfl_up(S, off);
    if (lane >= off) S = xcompose(o, S);
  }
  Xf G = xshfl_up(S, 1);  // exclusive prefix
  if (lane == 0) {
    G.cx = {1.f, 0.f, 0.f}; G.cy = {0.f, 1.f, 0.f}; G.cz = {0.f, 0.f, 1.f};
    G.t  = {0.f, 0.f, 0.f};
  }

  // ---- Pass 2: recompute local steps, transform by G, write out (NT) ----
  float* ob = out + (size_t)b * (4 * LRES * 3);
  if (lane == 0) {  // residue 0: seed atoms verbatim (G == identity)
    v4f q0 = {N0.x, N0.y, N0.z, CA0.x};
    v4f q1 = {CA0.y, CA0.z, C0.x, C0.y};
    v4f q2 = {C0.z, O0.x, O0.y, O0.z};
    __builtin_nontemporal_store(q0, (v4f*)(ob + 0));
    __builtin_nontemporal_store(q1, (v4f*)(ob + 4));
    __builtin_nontemporal_store(q2, (v4f*)(ob + 8));
  }
  N = N0; CA = CA0; C = C0;
  for (int s = 0; s < 16; ++s) {
    int i = base + s;
    if (i >= 0) {
      float tpsi = spsi[skew(i)], tomg = somg[skew(i)];
      float tphi = sphi[skew(i + 1)], to = sodh[skew(i)];
      F3 Nn  = place(N, CA, C, DBC_N, K_N, tpsi);
      F3 CAn = place(CA, C, Nn, DBC_CA, K_CA, tomg);
      F3 Cn  = place(C, Nn, CAn, DBC_C, K_C, tphi);
      F3 On  = place(Nn, CAn, Cn, DBC_O, K_O, to);
      N = Nn; CA = CAn; C = Cn;

      F3 gN = xapply(G, Nn), gCA = xapply(G, CAn);
      F3 gC = xapply(G, Cn), gO = xapply(G, On);
      float* p = ob + (size_t)(i + 1) * 12;  // residue r = i+1, atoms N,CA,C,O
      v4f q0 = {gN.x, gN.y, gN.z, gCA.x};
      v4f q1 = {gCA.y, gCA.z, gC.x, gC.y};
      v4f q2 = {gC.z, gO.x, gO.y, gO.z};
      __builtin_nontemporal_store(q0, (v4f*)(p + 0));
      __builtin_nontemporal_store(q1, (v4f*)(p + 4));
      __builtin_nontemporal_store(q2, (v4f*)(p + 8));
    }
  }
}

}  // namespace

extern "C" void kernel_launch(void* const* d_in, const int* in_sizes, int n_in,
                              void* d_out, int out_size, void* d_ws, size_t ws_size,
                              hipStream_t stream) {
  (void)n_in; (void)out_size; (void)d_ws; (void)ws_size;
  const float* phi = (const float*)d_in[0];
  const float* psi = (const float*)d_in[1];
  const float* omg = (const float*)d_in[2];
  const float* odh = (const float*)d_in[3];
  float* out = (float*)d_out;
  const int B = in_sizes[0] / LRES;  // 8192
  dim3 grid((B + WPB - 1) / WPB);
  nerf_build_kernel<<<grid, TPB, 0, stream>>>(phi, psi, omg, odh, out, B);
}